// JastrowSlater_4423816315385
// MI455X (gfx1250) — compile-verified
//
#include <hip/hip_runtime.h>
#include <math.h>

#define NSITES 256
#define NMODE  512
#define NEL    128
#define BATCH  4096
#define NB     16
#define LDAP   132   // padded LDS row stride in floats (multiple of 4 -> float4 aligned)

typedef float v2f __attribute__((ext_vector_type(2)));
typedef float v8f __attribute__((ext_vector_type(8)));

// ---------------------------------------------------------------------------
// Kernel A: Jastrow log factor.  log_j[b] = -0.5 * sum_{i,j occupied} V[i,j]
// One workgroup per batch element. V (1MB) is L2-resident; pure gather-sum.
// ---------------------------------------------------------------------------
__global__ __launch_bounds__(256) void jastrow_kernel(const int* __restrict__ n,
                                                      const float* __restrict__ v,
                                                      float* __restrict__ jl) {
    __shared__ int   occ[256];
    __shared__ float red[256];
    const int b   = blockIdx.x;
    const int tid = threadIdx.x;
    const int* nb_ = n + b * NMODE;
    if (tid == 0) {
        int k = 0;
        for (int t = 0; t < NMODE; ++t)
            if (nb_[t]) occ[k++] = t;          // exactly 256 occupied modes
    }
    __syncthreads();
    // thread tid owns fixed column occ[tid]; iterate over all occupied rows
    const int cj = occ[tid];
    float s = 0.0f;
#pragma unroll 4
    for (int i = 0; i < 256; ++i)
        s += v[occ[i] * NMODE + cj];
    red[tid] = s;
    __syncthreads();
    for (int st = 128; st > 0; st >>= 1) {
        if (tid < st) red[tid] += red[tid + st];
        __syncthreads();
    }
    if (tid == 0) jl[b] = -0.5f * red[0];
}

// ---------------------------------------------------------------------------
// Kernel B: slogdet of gathered 128x128 matrix via blocked LU (NB=16) with
// partial pivoting, entirely in LDS. Trailing update A22 -= L21*U12 runs on
// the matrix core: V_WMMA_F32_16X16X4_F32, K=16 as 4 chained K=4 WMMAs.
// grid = 8192 workgroups: g = spin*4096 + b.
// ---------------------------------------------------------------------------
__global__ __launch_bounds__(256) void slogdet_kernel(const int* __restrict__ n,
                                                      const float* __restrict__ phi_up,
                                                      const float* __restrict__ phi_dn,
                                                      float* __restrict__ sgn_out,
                                                      float* __restrict__ ld_out) {
    extern __shared__ char smem_raw[];
    float* A    = (float*)smem_raw;             // NEL * LDAP floats
    int*   idx  = (int*)(A + NEL * LDAP);       // 128 occupied site indices
    float* rv   = (float*)(idx + NEL);          // pivot reduction values
    int*   ri   = (int*)(rv + NEL);             // pivot reduction indices
    float* acc  = (float*)(ri + NEL);           // [0]=sign  [1]=log|det|
    int*   pivs = (int*)(acc + 2);              // selected pivot row

    const int g    = blockIdx.x;
    const int b    = g & (BATCH - 1);
    const int spin = g >> 12;
    const int tid  = threadIdx.x;
    const int* nb_ = n + b * NMODE + spin * NSITES;
    const float* phi = spin ? phi_dn : phi_up;

    if (tid == 0) {
        int k = 0;
        for (int t = 0; t < NSITES; ++t)
            if (nb_[t]) idx[k++] = t;           // ascending occupied indices (exactly 128)
        acc[0] = 1.0f;
        acc[1] = 0.0f;
    }
    __syncthreads();

    // Gather phi rows into LDS (float4; LDAP*4 bytes = 528 is 16B-aligned)
    const float4* ph4 = (const float4*)phi;
    for (int e = tid; e < NEL * (NEL / 4); e += 256) {
        const int r  = e >> 5;                  // 32 float4 per 128-float row
        const int c4 = e & 31;
        float4 val = ph4[idx[r] * (NEL / 4) + c4];
        *(float4*)(&A[r * LDAP + 4 * c4]) = val;
    }
    __syncthreads();

    for (int kb = 0; kb < NEL / NB; ++kb) {
        const int j0 = kb * NB;

        // ----- panel factorization: columns [j0, j0+NB), partial pivoting -----
        for (int j = j0; j < j0 + NB; ++j) {
            const int cnt = NEL - j;
            if (tid < 128) {
                if (tid < cnt) { rv[tid] = fabsf(A[(j + tid) * LDAP + j]); ri[tid] = j + tid; }
                else           { rv[tid] = -1.0f;                          ri[tid] = j; }
            }
            __syncthreads();
            for (int st = 64; st > 0; st >>= 1) {
                if (tid < st) {
                    if (rv[tid + st] > rv[tid]) { rv[tid] = rv[tid + st]; ri[tid] = ri[tid + st]; }
                }
                __syncthreads();
            }
            if (tid == 0) {
                const int p = ri[0];
                pivs[0] = p;
                if (p != j) acc[0] = -acc[0];   // row-swap parity
            }
            __syncthreads();
            const int p = pivs[0];
            if (p != j && tid < NEL) {          // swap full rows j <-> p
                const float t0 = A[j * LDAP + tid];
                A[j * LDAP + tid] = A[p * LDAP + tid];
                A[p * LDAP + tid] = t0;
            }
            __syncthreads();
            const float diag = A[j * LDAP + j];
            if (tid == 0) {
                acc[0] *= (diag > 0.0f) ? 1.0f : ((diag < 0.0f) ? -1.0f : 0.0f);
                acc[1] += logf(fabsf(diag));
            }
            const float inv = 1.0f / diag;
            const int h = NEL - 1 - j;
            if (tid < h) {                      // scale column + rank-1 update inside panel
                const int i = j + 1 + tid;
                const float lij = A[i * LDAP + j] * inv;
                A[i * LDAP + j] = lij;
                for (int jj = j + 1; jj < j0 + NB; ++jj)
                    A[i * LDAP + jj] -= lij * A[j * LDAP + jj];
            }
            __syncthreads();
        }

        const int c0r = j0 + NB;
        const int rem = NEL - c0r;
        if (rem > 0) {
            // ----- U12 = L11^{-1} * A12 : each thread owns one trailing column,
            // all dependencies stay within the column -> no barriers needed -----
            if (tid < rem) {
                const int c = c0r + tid;
#pragma unroll
                for (int i = 1; i < NB; ++i) {
                    float a = A[(j0 + i) * LDAP + c];
                    for (int m = 0; m < i; ++m)
                        a -= A[(j0 + i) * LDAP + (j0 + m)] * A[(j0 + m) * LDAP + c];
                    A[(j0 + i) * LDAP + c] = a;
                }
            }
            __syncthreads();

            // ----- trailing GEMM on matrix core: A22 -= L21 * U12 -----
            const int nt    = rem >> 4;                 // 16x16 tiles per side
            const int wave  = tid >> 5;
            const int lane  = tid & 31;
            const int nlo   = lane & 15;                // M (A-frag) / N (B,C-frag) sub-index
            const int khalf = (lane >> 4) << 1;         // 0 or 2  (K pair select)
            const int mhi   = (lane >> 4) << 3;         // 0 or 8  (C/D row-half select)
            for (int tile = wave; tile < nt * nt; tile += 8) {
                const int tr = tile / nt, tc = tile - tr * nt;
                const int r0 = c0r + tr * 16, c0 = c0r + tc * 16;
                v8f cfrag;
#pragma unroll
                for (int vv = 0; vv < 8; ++vv)          // C/D layout: VGPR vv -> M = vv (+8)
                    cfrag[vv] = A[(r0 + vv + mhi) * LDAP + c0 + nlo];
#pragma unroll
                for (int kk = 0; kk < NB; kk += 4) {    // K=16 as 4 x (K=4) WMMA
                    const int kcol = j0 + kk;
                    v2f afrag, bfrag;
                    // A 16x4: lane<16 -> K={0,1}, lane>=16 -> K={2,3}; negate for subtraction
                    afrag.x = -A[(r0 + nlo) * LDAP + kcol + khalf];
                    afrag.y = -A[(r0 + nlo) * LDAP + kcol + khalf + 1];
                    // B 4x16: mirrored K striping, N striped across lanes
                    bfrag.x =  A[(kcol + khalf)     * LDAP + c0 + nlo];
                    bfrag.y =  A[(kcol + khalf + 1) * LDAP + c0 + nlo];
                    cfrag = __builtin_amdgcn_wmma_f32_16x16x4_f32(
                        false, afrag, false, bfrag, (short)0, cfrag, false, false);
                }
#pragma unroll
                for (int vv = 0; vv < 8; ++vv)
                    A[(r0 + vv + mhi) * LDAP + c0 + nlo] = cfrag[vv];
            }
            __syncthreads();
        }
    }

    if (tid == 0) {
        sgn_out[g] = acc[0];
        ld_out[g]  = acc[1];
    }
}

// ---------------------------------------------------------------------------
// Kernel C: combine -> out[0][b] = sign_up*sign_dn ; out[1][b] = jl + ld_up + ld_dn
// ---------------------------------------------------------------------------
__global__ __launch_bounds__(256) void combine_kernel(const float* __restrict__ sgn,
                                                      const float* __restrict__ ld,
                                                      const float* __restrict__ jl,
                                                      float* __restrict__ out) {
    const int b = blockIdx.x * blockDim.x + threadIdx.x;
    if (b < BATCH) {
        out[b]         = sgn[b] * sgn[BATCH + b];
        out[BATCH + b] = jl[b] + ld[b] + ld[BATCH + b];
    }
}

extern "C" void kernel_launch(void* const* d_in, const int* in_sizes, int n_in,
                              void* d_out, int out_size, void* d_ws, size_t ws_size,
                              hipStream_t stream) {
    const int*   n      = (const int*)d_in[0];
    const float* phi_up = (const float*)d_in[1];
    const float* phi_dn = (const float*)d_in[2];
    const float* v      = (const float*)d_in[3];
    float* out = (float*)d_out;

    float* ws  = (float*)d_ws;
    float* sgn = ws;               // 8192 floats  (up: [0,4096), dn: [4096,8192))
    float* ld  = ws + 2 * BATCH;   // 8192 floats
    float* jl  = ws + 4 * BATCH;   // 4096 floats

    const size_t smem = (size_t)(NEL * LDAP + 3 * NEL) * sizeof(float) + 16;

    jastrow_kernel<<<BATCH, 256, 0, stream>>>(n, v, jl);
    slogdet_kernel<<<2 * BATCH, 256, smem, stream>>>(n, phi_up, phi_dn, sgn, ld);
    combine_kernel<<<(BATCH + 255) / 256, 256, 0, stream>>>(sgn, ld, jl, out);
}